// DecisionTrees2_74792560493296
// MI455X (gfx1250) — compile-verified
//
#include <hip/hip_runtime.h>

// CDNA5 / gfx1250, wave32.
// DecisionTrees2: soft-binning (8 features x 4 bins) followed by an 8-way
// Kronecker product -> 2048 x 65536 f32 output (512 MB).
//
// Roofline: store-bandwidth bound. 512MB / 23.3 TB/s ~= 22us floor; compute
// (0.5M rank-1 WMMAs + ~0.13 GFLOP of VALU) is negligible. Strategy:
//   P_hi[256] (features 0..3) (x) P_lo[256] (features 4..7)
// expanded with V_WMMA_F32_16X16X4_F32 (K=1 rank-1 outer product, full f32).
// All WMMA operands are preloaded into registers before the tile loop
// (zero-padded LDS regions kill the exec-divergent conditional ds_loads seen
// in the previous build), so the inner loop is pure WMMA + NT store clauses.

typedef __attribute__((ext_vector_type(2))) float v2f;
typedef __attribute__((ext_vector_type(4))) float v4f;
typedef __attribute__((ext_vector_type(8))) float v8f;

__global__ __launch_bounds__(256) void decision_trees2_kernel(
    const float* __restrict__ x,          // [N, 8]
    const float* __restrict__ cut_points, // [8, 3]
    float* __restrict__ out)              // [N, 65536]
{
    __shared__ float s_bins[32];                   // [feature*4 + bin]
    __shared__ float s_hi[512];                    // [0..255]=P_hi, [256..511]=0
    __shared__ __align__(16) float s_loT[32][20];  // P_lo transposed: [col][tl];
                                                   // rows 16..31 = 0; stride 20
                                                   // (rows 16B-aligned, lane->bank
                                                   //  stride 20 ~conflict-free)

    const int tid = threadIdx.x;
    const int n   = blockIdx.x;

    // ---- Phase 1: per-feature soft binning (threads 0..7, one per feature)
    if (tid < 8) {
        const int f = tid;
        float c0 = cut_points[f * 3 + 0];
        float c1 = cut_points[f * 3 + 1];
        float c2 = cut_points[f * 3 + 2];
        float t;
        if (c0 > c1) { t = c0; c0 = c1; c1 = t; }
        if (c0 > c2) { t = c0; c0 = c2; c2 = t; }
        if (c1 > c2) { t = c1; c1 = c2; c2 = t; }
        const float b1 = -c0;
        const float b2 = -c0 - c1;
        const float b3 = -c0 - c1 - c2;
        const float xv = x[n * 8 + f];
        const float h0 = xv * 1.0f;               // W = [1,2,3,4]
        const float h1 = xv * 2.0f + b1;
        const float h2 = xv * 3.0f + b2;
        const float h3 = xv * 4.0f + b3;
        const float m  = fmaxf(fmaxf(h0, h1), fmaxf(h2, h3));
        const float e0 = expf((h0 - m) * 10.0f);  // /T, T=0.1
        const float e1 = expf((h1 - m) * 10.0f);
        const float e2 = expf((h2 - m) * 10.0f);
        const float e3 = expf((h3 - m) * 10.0f);
        const float inv = 1.0f / (e0 + e1 + e2 + e3);
        s_bins[f * 4 + 0] = e0 * inv;
        s_bins[f * 4 + 1] = e1 * inv;
        s_bins[f * 4 + 2] = e2 * inv;
        s_bins[f * 4 + 3] = e3 * inv;
    }
    __syncthreads();

    // ---- Phase 2: partial products + zero padding.
    // j = d0*4^7+...+d7 ; hi = j>>8 (f0..f3, f0 most significant), lo = j&255.
    {
        const int t = tid;
        const float ph = s_bins[ 0 + ((t >> 6) & 3)] *
                         s_bins[ 4 + ((t >> 4) & 3)] *
                         s_bins[ 8 + ((t >> 2) & 3)] *
                         s_bins[12 + ( t       & 3)];
        const float pl = s_bins[16 + ((t >> 6) & 3)] *
                         s_bins[20 + ((t >> 4) & 3)] *
                         s_bins[24 + ((t >> 2) & 3)] *
                         s_bins[28 + ( t       & 3)];
        s_hi[t]       = ph;
        s_hi[256 + t] = 0.0f;                       // zero region for lanes 16..31
        s_loT[(t & 15)][t >> 4]      = pl;          // P_lo[(t>>4)*16 + (t&15)] = P_lo[t]
        s_loT[16 + (t & 15)][t >> 4] = 0.0f;        // zero rows for lanes 16..31
    }
    __syncthreads();

    // ---- Phase 3: rank-1 outer products on the matrix unit.
    // A (16x4 f32): lanes 0-15 {K0,K1}, lanes 16-31 {K2,K3} -> only (lane<16, v0)=K0 used.
    // B (4x16 f32): row K across lanes in one VGPR -> only (lane<16, v0)=K0 row used.
    // D (16x16 f32): VGPR r: lanes 0-15 -> row r, lanes 16-31 -> row r+8; col = lane&15.
    const int wave  = tid >> 5;
    const int lane  = tid & 31;
    const int col   = lane & 15;
    const int zoff  = (lane >> 4) << 8;   // 0 for lanes 0-15, 256 (zero region) else
    const int mbase = (lane >> 4) << 3;   // D row offset for upper lane half

    // Preload all 16 B fragments (zeros for lanes 16..31) -> 4x ds_load_b128.
    const v4f* rowp = (const v4f*)(&s_loT[lane][0]);
    const v4f q0 = rowp[0], q1 = rowp[1], q2 = rowp[2], q3 = rowp[3];
    const float bl[16] = {q0.x, q0.y, q0.z, q0.w,  q1.x, q1.y, q1.z, q1.w,
                          q2.x, q2.y, q2.z, q2.w,  q3.x, q3.y, q3.z, q3.w};

    // Preload the 2 A fragments this wave owns.
    const int   th0 = wave * 2;
    const float a0  = s_hi[zoff + th0 * 16 + col];
    const float a1  = s_hi[zoff + (th0 + 1) * 16 + col];

    float* const obase = out + (size_t)n * 65536;

    #pragma unroll
    for (int i = 0; i < 2; ++i) {
        v2f a;
        a.x = (i == 0) ? a0 : a1;   // K=0 column of A
        a.y = 0.0f;
        float* const pbase = obase + (size_t)((th0 + i) * 16 + mbase) * 256 + col;
        #pragma unroll
        for (int tl = 0; tl < 16; ++tl) {
            v2f b;
            b.x = bl[tl];           // K=0 row of B
            b.y = 0.0f;
            v8f c = {};
            // D[m][ncol] = P_hi[(th0+i)*16+m] * P_lo[tl*16+ncol]
            c = __builtin_amdgcn_wmma_f32_16x16x4_f32(
                    /*neg_a=*/false, a, /*neg_b=*/false, b,
                    /*c_mod=*/(short)0, c, /*reuse_a=*/false, /*reuse_b=*/false);
            // Streaming output (512MB >> 192MB L2) -> non-temporal stores.
            float* p = pbase + tl * 16;
            #pragma unroll
            for (int r = 0; r < 8; ++r) {
                __builtin_nontemporal_store(c[r], p + (size_t)r * 256);
            }
        }
    }
}

extern "C" void kernel_launch(void* const* d_in, const int* in_sizes, int n_in,
                              void* d_out, int out_size, void* d_ws, size_t ws_size,
                              hipStream_t stream) {
    const float* x   = (const float*)d_in[0];   // (N, 8) f32
    const float* cps = (const float*)d_in[1];   // (8, 3) f32
    float* out = (float*)d_out;                 // (N, 65536) f32
    const int N = in_sizes[0] / 8;              // 2048
    decision_trees2_kernel<<<N, 256, 0, stream>>>(x, cps, out);
}